// DynamicKVCompressor_75376676045039
// MI455X (gfx1250) — compile-verified
//
#include <hip/hip_runtime.h>

// ---------------------------------------------------------------------------
// DynamicKVCompressor for MI455X (gfx1250, wave32)
//
// Shapes (static): L=4 H=8 B=2 S=2048 D=128 W=128 K=512
//   attn   [L,H,B,S,S] f32
//   keys   [L,B,S,H,D] f32
//   values [L,B,S,H,D] f32
// Outputs (concat in d_out, f32):
//   compressed_keys   [L,B,K,H,D]  = 4,194,304
//   compressed_values [L,B,K,H,D]  = 4,194,304
//   keep_idx          [L,B,K]      = 4,096 (indices as float)
//
// Key simplification: the per-layer max-normalization is a positive scalar
// scale -> top-k ordering invariant -> skipped (normalized scores are not
// part of the outputs).
// ---------------------------------------------------------------------------

#define NL 4
#define NH 8
#define NB 2
#define SQ 2048
#define HD 128
#define WIN 128
#define TOPK 512

typedef __attribute__((ext_vector_type(2))) float v2f;
typedef __attribute__((ext_vector_type(8))) float v8f;

// ---------------------------------------------------------------------------
// Phase 1: windowed attention score reduction on the matrix pipe.
// scores[l,b,sk] = sum over 1024 rows r=(h,qw) of attn[l,h,b,1920+qw,sk].
// One wave per 16-column sk tile: D = ones(16x4) x attnRows(4x16) + C,
// chained over 256 chunks of 4 rows. All 16 rows of D are the column sums;
// lanes 0..15, acc[0] hold scores[col0..col0+15].
// ---------------------------------------------------------------------------
__global__ void __launch_bounds__(256)
kv_scores_wmma(const float* __restrict__ attn, float* __restrict__ scores) {
    const int wave = threadIdx.x >> 5;
    const int lane = threadIdx.x & 31;
    const int tile = blockIdx.x * 8 + wave;      // 0..1023 = L*B*(S/16)
    const int l    = tile >> 8;                  // / (B * 128)
    const int b    = (tile >> 7) & 1;
    const int col  = ((tile & 127) << 4) + (lane & 15);
    const int half = lane >> 4;                  // lanes 0-15: rows K=0,1 ; 16-31: K=2,3

    const v2f ones = {1.0f, 1.0f};
    v8f acc = {};

    for (int h = 0; h < NH; ++h) {
        // row r = h*WIN + qw ; this lane starts at qw = 2*half, q = 1920+qw
        const float* __restrict__ p = attn
            + ((((size_t)(l * NH + h) * NB + b) * SQ + (SQ - WIN) + 2 * half) * SQ + col);
        #pragma unroll 8
        for (int c = 0; c < WIN / 4; ++c) {      // 32 chunks of 4 q-rows
            v2f bm;
            bm[0] = p[0];                        // row K = 2*half
            bm[1] = p[SQ];                       // row K = 2*half + 1
            acc = __builtin_amdgcn_wmma_f32_16x16x4_f32(
                false, ones, false, bm, (short)0, acc, false, false);
            p += 4 * SQ;
        }
    }

    if (lane < 16) {
        scores[((l * NB + b) * SQ) + col] = acc[0];  // M=0 row of D
    }
}

// ---------------------------------------------------------------------------
// Phase 2: stable top-512 of 2048 per (l,b) by rank counting.
// rank(i) = #{ j : s[j] > s[i]  ||  (s[j]==s[i] && j<i) }   (strict total order)
// keep iff rank < 512  -> exactly 512 kept, ties resolved like lax.top_k.
// Prefix scan of the keep mask emits indices ascending (== jnp.sort(top_idx)).
// ---------------------------------------------------------------------------
__global__ void __launch_bounds__(256)
kv_topk(const float* __restrict__ scores, int* __restrict__ keep,
        float* __restrict__ keep_f) {
    __shared__ float s[SQ];
    __shared__ int cnt[256];

    const int lb  = blockIdx.x;                  // 0..7 = L*B
    const int tid = threadIdx.x;
    const float* __restrict__ sc = scores + (size_t)lb * SQ;

    for (int i = tid; i < SQ; i += 256) s[i] = sc[i];
    __syncthreads();

    const int base = tid * 8;                    // 8 contiguous elements/thread
    float v[8];
    int   rank[8];
    #pragma unroll
    for (int e = 0; e < 8; ++e) { v[e] = s[base + e]; rank[e] = 0; }

    for (int j = 0; j < SQ; ++j) {
        const float u = s[j];
        #pragma unroll
        for (int e = 0; e < 8; ++e)
            rank[e] += (u > v[e]) | ((u == v[e]) & (j < base + e));
    }

    int keepmask = 0, local = 0;
    #pragma unroll
    for (int e = 0; e < 8; ++e)
        if (rank[e] < TOPK) { keepmask |= (1 << e); ++local; }

    cnt[tid] = local;
    __syncthreads();
    // Hillis-Steele inclusive scan over 256 per-thread counts
    for (int off = 1; off < 256; off <<= 1) {
        int t = (tid >= off) ? cnt[tid - off] : 0;
        __syncthreads();
        if (tid >= off) cnt[tid] += t;
        __syncthreads();
    }

    int pos = (size_t)lb * TOPK + (cnt[tid] - local);  // exclusive prefix
    #pragma unroll
    for (int e = 0; e < 8; ++e) {
        if (keepmask & (1 << e)) {
            const int i = base + e;
            keep[pos]   = i;
            keep_f[pos] = (float)i;               // keep_idx output (as f32)
            ++pos;
        }
    }
}

// ---------------------------------------------------------------------------
// Phase 3: gather kept K/V rows. One block per (l,b,k) row of H*D = 1024
// floats; 256 threads x float4 -> global_load_b128 / global_store_b128.
// ---------------------------------------------------------------------------
__global__ void __launch_bounds__(256)
kv_gather(const float* __restrict__ keys, const float* __restrict__ values,
          const int* __restrict__ keep,
          float* __restrict__ outk, float* __restrict__ outv) {
    const int n   = blockIdx.x;                  // 0..4095 = L*B*K
    const int lb  = n >> 9;
    const int idx = keep[n];

    const size_t src = ((size_t)lb * SQ + idx) * (NH * HD);
    const size_t dst = (size_t)n * (NH * HD);

    const float4* __restrict__ ks = (const float4*)(keys   + src);
    const float4* __restrict__ vs = (const float4*)(values + src);
    float4* __restrict__ kd = (float4*)(outk + dst);
    float4* __restrict__ vd = (float4*)(outv + dst);

    kd[threadIdx.x] = ks[threadIdx.x];           // 256 * 16B = 4 KiB row
    vd[threadIdx.x] = vs[threadIdx.x];
}

// ---------------------------------------------------------------------------
extern "C" void kernel_launch(void* const* d_in, const int* in_sizes, int n_in,
                              void* d_out, int out_size, void* d_ws, size_t ws_size,
                              hipStream_t stream) {
    const float* attn   = (const float*)d_in[0];
    const float* keys   = (const float*)d_in[1];
    const float* values = (const float*)d_in[2];

    const size_t kv_elems = (size_t)NL * NB * TOPK * NH * HD;   // 4,194,304
    float* outk   = (float*)d_out;
    float* outv   = outk + kv_elems;
    float* outidx = outv + kv_elems;                            // 4096 floats

    float* scores = (float*)d_ws;                               // 16384 f32
    int*   keep   = (int*)(scores + (size_t)NL * NB * SQ);      // 4096 i32

    // Phase 1: 1024 sk-tiles, 8 waves/block -> 128 blocks. Reads 64 MiB.
    kv_scores_wmma<<<128, 256, 0, stream>>>(attn, scores);
    // Phase 2: one block per (l,b).
    kv_topk<<<NL * NB, 256, 0, stream>>>(scores, keep, outidx);
    // Phase 3: one block per kept row. Moves 64 MiB.
    kv_gather<<<NL * NB * TOPK, 256, 0, stream>>>(keys, values, keep, outk, outv);
}